// SetConvEncoder_30872224924407
// MI455X (gfx1250) — compile-verified
//
#include <hip/hip_runtime.h>
#include <math.h>

typedef __attribute__((ext_vector_type(2))) float v2f;
typedef __attribute__((ext_vector_type(8))) float v8f;

#define M_BATCH 8
#define NC 1024
#define NT 512
#define GS 129                     // grid side = 2*64+1
#define NG (GS*GS)                 // 16641 grid points
#define NHALF 64
#define XGRID_FLOATS (M_BATCH*NG*2)
#define NTILES ((NG + 15)/16)      // 1041 tiles of 16 grid points

// ---------------------------------------------------------------------------
// Kernel 1: per-batch bounding box midpoints + inverse lengthscale -> ws
// ws[0..1] = 1/lengthscale[d]; ws[2+2m+d] = x_mid[m][d]
// ---------------------------------------------------------------------------
__global__ __launch_bounds__(256)
void prep_kernel(const float* __restrict__ xc, const float* __restrict__ xt,
                 const float* __restrict__ lsp, float* __restrict__ ws) {
    const int m = blockIdx.x;
    const int tid = threadIdx.x;
    __shared__ float rmn0[256], rmx0[256], rmn1[256], rmx1[256];
    float mn0 = 1e30f, mx0 = -1e30f, mn1 = 1e30f, mx1 = -1e30f;
    for (int p = tid; p < NC + NT; p += 256) {
        float v0, v1;
        if (p < NC) { v0 = xc[(m*NC + p)*2 + 0]; v1 = xc[(m*NC + p)*2 + 1]; }
        else { int q = p - NC; v0 = xt[(m*NT + q)*2 + 0]; v1 = xt[(m*NT + q)*2 + 1]; }
        mn0 = fminf(mn0, v0); mx0 = fmaxf(mx0, v0);
        mn1 = fminf(mn1, v1); mx1 = fmaxf(mx1, v1);
    }
    rmn0[tid] = mn0; rmx0[tid] = mx0; rmn1[tid] = mn1; rmx1[tid] = mx1;
    __syncthreads();
    if (tid == 0) {
        for (int i = 1; i < 256; ++i) {
            mn0 = fminf(mn0, rmn0[i]); mx0 = fmaxf(mx0, rmx0[i]);
            mn1 = fminf(mn1, rmn1[i]); mx1 = fmaxf(mx1, rmx1[i]);
        }
        ws[2 + m*2 + 0] = 0.5f*(mn0 + mx0);
        ws[2 + m*2 + 1] = 0.5f*(mn1 + mx1);
        if (m == 0) {
            for (int d = 0; d < 2; ++d) {
                float ls = 1e-5f + log1pf(__expf(lsp[d]));   // softplus
                ws[d] = 1.0f / ls;
            }
        }
    }
}

// ---------------------------------------------------------------------------
// Kernel 2: x_grid fill: out[(m*NG+g)*2+d] = mid[d] + (idx-64)/64
// ---------------------------------------------------------------------------
__global__ __launch_bounds__(256)
void xgrid_kernel(const float* __restrict__ ws, float* __restrict__ out) {
    const int idx = blockIdx.x*256 + threadIdx.x;
    if (idx >= M_BATCH*NG) return;
    const int m = idx / NG, g = idx % NG;
    const int i = g / GS, j = g % GS;
    out[idx*2 + 0] = ws[2 + m*2 + 0] + (float)(i - NHALF) * (1.0f/64.0f);
    out[idx*2 + 1] = ws[2 + m*2 + 1] + (float)(j - NHALF) * (1.0f/64.0f);
}

// ---------------------------------------------------------------------------
// Kernel 3: z = exp(-0.5*dist2) @ [yc,1]
// Cross term a.b via V_WMMA_F32_16X16X4_F32 (dim=2 padded to K=4).
// A = 16 context pts (M) x K, B = K x 16 grid pts (N).
// D layout: lane owns grid col N=lane%16; ctx rows striped over 8 acc VGPRs
// (rows +8 for lanes>=16) -> z accumulation is lane-local.
// ---------------------------------------------------------------------------
__global__ __launch_bounds__(256)
void setconv_kernel(const float* __restrict__ xc, const float* __restrict__ yc,
                    const float* __restrict__ ws, float* __restrict__ zout) {
    __shared__ __align__(16) float s_b4[NC*4];  // (bx,by,0,0) scaled ctx, 16KB
    __shared__ __align__(16) float s_bn[NC];    // |b|^2, 4KB
    __shared__ __align__(16) float s_y[NC*2];   // yc channels, 8KB
    const int mb  = blockIdx.y;
    const int tid = threadIdx.x;
    const float ils0 = ws[0], ils1 = ws[1];
    const float mid0 = ws[2 + mb*2 + 0], mid1 = ws[2 + mb*2 + 1];

    for (int c = tid; c < NC; c += 256) {
        float bx = xc[(mb*NC + c)*2 + 0] * ils0;
        float by = xc[(mb*NC + c)*2 + 1] * ils1;
        s_b4[c*4+0] = bx; s_b4[c*4+1] = by; s_b4[c*4+2] = 0.f; s_b4[c*4+3] = 0.f;
        s_bn[c] = bx*bx + by*by;
        s_y[c*2+0] = yc[(mb*NC + c)*2 + 0];
        s_y[c*2+1] = yc[(mb*NC + c)*2 + 1];
    }
    __syncthreads();

    const int lane   = tid & 31;
    const int wave   = tid >> 5;
    const int lane16 = lane & 15;
    const bool hi    = lane >= 16;
    const int rowoff = hi ? 8 : 0;

    for (int t = blockIdx.x*8 + wave; t < NTILES; t += gridDim.x*8) {
        const int g  = t*16 + lane16;
        const int gc = min(g, NG - 1);
        const int gi = gc / GS, gj = gc % GS;
        const float ax = (mid0 + (float)(gi - NHALF)*(1.0f/64.0f)) * ils0;
        const float ay = (mid1 + (float)(gj - NHALF)*(1.0f/64.0f)) * ils1;
        const float anorm = ax*ax + ay*ay;
        v2f bop; bop[0] = hi ? 0.f : ax; bop[1] = hi ? 0.f : ay;  // zeros fill K=2,3 rows
        float z0 = 0.f, z1 = 0.f, z2 = 0.f;

        for (int base = 0; base < NC; base += 16) {
            // A operand: lanes<16 fetch (bx,by) = K0,K1; lanes>=16 fetch (0,0) = K2,K3
            const float2 av = *(const float2*)&s_b4[(base + lane16)*4 + (hi ? 2 : 0)];
            v2f aop; aop[0] = av.x; aop[1] = av.y;
            v8f acc = {};
            acc = __builtin_amdgcn_wmma_f32_16x16x4_f32(
                false, aop, false, bop, (short)0, acc, false, false);

            const float4 bn0 = *(const float4*)&s_bn[base + rowoff];
            const float4 bn1 = *(const float4*)&s_bn[base + rowoff + 4];
            const float4 y01 = *(const float4*)&s_y[(base + rowoff)*2 + 0];
            const float4 y23 = *(const float4*)&s_y[(base + rowoff)*2 + 4];
            const float4 y45 = *(const float4*)&s_y[(base + rowoff)*2 + 8];
            const float4 y67 = *(const float4*)&s_y[(base + rowoff)*2 + 12];
            const float bn[8] = {bn0.x,bn0.y,bn0.z,bn0.w, bn1.x,bn1.y,bn1.z,bn1.w};
            const float yv[16] = {y01.x,y01.y,y01.z,y01.w, y23.x,y23.y,y23.z,y23.w,
                                  y45.x,y45.y,y45.z,y45.w, y67.x,y67.y,y67.z,y67.w};
            #pragma unroll
            for (int r = 0; r < 8; ++r) {
                float d2 = fmaf(-2.0f, acc[r], bn[r] + anorm);
                d2 = fmaxf(d2, 0.0f);
                float w = __expf(-0.5f * d2);       // v_exp_f32, co-executes with XDL
                z0 = fmaf(w, yv[2*r+0], z0);
                z1 = fmaf(w, yv[2*r+1], z1);
                z2 += w;                            // density channel (y==1)
            }
        }
        // merge the two lane halves (ctx rows 0-7 vs 8-15)
        z0 += __shfl_xor(z0, 16, 32);
        z1 += __shfl_xor(z1, 16, 32);
        z2 += __shfl_xor(z2, 16, 32);
        if (!hi && g < NG) {
            float* p = zout + (size_t)(mb*NG + g)*3;
            p[0] = z0; p[1] = z1; p[2] = z2;
        }
    }
}

// ---------------------------------------------------------------------------
extern "C" void kernel_launch(void* const* d_in, const int* in_sizes, int n_in,
                              void* d_out, int out_size, void* d_ws, size_t ws_size,
                              hipStream_t stream) {
    const float* xc  = (const float*)d_in[0];
    const float* yc  = (const float*)d_in[1];
    const float* xt  = (const float*)d_in[2];
    const float* lsp = (const float*)d_in[3];
    float* out = (float*)d_out;
    float* ws  = (float*)d_ws;

    prep_kernel<<<M_BATCH, 256, 0, stream>>>(xc, xt, lsp, ws);

    const int total = M_BATCH * NG;
    xgrid_kernel<<<(total + 255)/256, 256, 0, stream>>>(ws, out);

    dim3 grid(33, M_BATCH);   // 33*8 = 264 waves per batch over 1041 tiles
    setconv_kernel<<<grid, 256, 0, stream>>>(xc, yc, ws, out + XGRID_FLOATS);
}